// BaseQuantizer_17995912970288
// MI455X (gfx1250) — compile-verified
//
#include <hip/hip_runtime.h>

typedef float v2f __attribute__((ext_vector_type(2)));
typedef float v8f __attribute__((ext_vector_type(8)));

#define B_   64
#define C_   32
#define H_   16
#define W_   16
#define K_   4096
#define TOT  (B_ * C_ * H_ * W_)   // 524288

// ---------------------------------------------------------------- init
__global__ void init_state(const float* __restrict__ z_enc,
                           float* __restrict__ z_rest,
                           float* __restrict__ z_dec) {
    int i = blockIdx.x * blockDim.x + threadIdx.x;
    if (i < TOT) { z_rest[i] = z_enc[i]; z_dec[i] = 0.f; }
}

// 0.5 * ||e_n||^2  (argmin dist == argmax (z.e - 0.5||e||^2))
__global__ void wsq_half(const float* __restrict__ emb, float* __restrict__ wsqh) {
    int n = blockIdx.x * blockDim.x + threadIdx.x;
    if (n < K_) {
        float s = 0.f;
#pragma unroll
        for (int c = 0; c < C_; ++c) { float v = emb[n * C_ + c]; s += v * v; }
        wsqh[n] = 0.5f * s;
    }
}

// ---------------------------------------------------------------- argmin (WMMA)
// One 128-thread block = 4 waves; each wave owns one 16-row tile of zflat and
// scans all 4096 codes in 16-column tiles: 8x V_WMMA_F32_16X16X4_F32 per tile
// (K=32 split into 8 chunks of 4).  Exact f32 math = reference precision.
__global__ __launch_bounds__(128)
void vq_argmin(const float* __restrict__ z_rest, const float* __restrict__ emb,
               const float* __restrict__ wsqh, int* __restrict__ tokens, int ph) {
    __shared__ float redS[4][16][16];
    __shared__ int   redN[4][16][16];

    const int wave = threadIdx.x >> 5;
    const int lane = threadIdx.x & 31;
    const int m    = lane & 15;        // A row / C column index
    const int grp  = lane >> 4;        // 0 or 1 (selects K pair / M half)
    const int kb   = grp * 2;          // K offset within 4-chunk (A & B layout)

    const int tile = blockIdx.x * 4 + wave;
    const int row  = tile * 16 + m;    // zflat row = b*ph*ph + y*ph + x
    const int pp   = ph * ph;
    const int b    = row / pp;
    const int rem  = row - b * pp;
    const int y    = rem / ph;
    const int x    = rem - y * ph;
    const int s    = 16 / ph;          // pooling block edge (1 for last level)
    const float inv = 1.0f / (float)(s * s);

    // Preload A: pooled zflat[row][k] for the 16 channels this lane feeds.
    v2f a[8];
#pragma unroll
    for (int kk = 0; kk < 8; ++kk) {
        const int c0 = kk * 4 + kb;
        const float* p0 = z_rest + ((b * C_ + c0) * H_ + y * s) * W_ + x * s;
        const float* p1 = p0 + H_ * W_;          // channel c0+1
        float v0 = 0.f, v1 = 0.f;
        for (int i2 = 0; i2 < s; ++i2)
            for (int j2 = 0; j2 < s; ++j2) {
                v0 += p0[i2 * W_ + j2];
                v1 += p1[i2 * W_ + j2];
            }
        a[kk].x = v0 * inv;
        a[kk].y = v1 * inv;
    }

    float bs[8];
    int   bn[8];
#pragma unroll
    for (int r = 0; r < 8; ++r) { bs[r] = -3.4e38f; bn[r] = 0; }

    for (int nb = 0; nb < K_ / 16; ++nb) {
        const int n = nb * 16 + m;                 // this lane's B column
        const float* bp = emb + n * C_ + kb;
        if (nb + 1 < K_ / 16)                      // pull next code tile into L1
            __builtin_prefetch(bp + 16 * C_, 0, 0);
        v8f acc = {};
#pragma unroll
        for (int kk = 0; kk < 8; ++kk) {
            v2f bv = *(const v2f*)(bp + kk * 4);
            acc = __builtin_amdgcn_wmma_f32_16x16x4_f32(
                      false, a[kk], false, bv, (short)0, acc, false, false);
        }
        const float wq = wsqh[n];
#pragma unroll
        for (int r = 0; r < 8; ++r) {
            float sc = acc[r] - wq;                // score for (M=r+8*grp, N=n)
            if (sc > bs[r]) { bs[r] = sc; bn[r] = n; }
        }
    }

    // cross-lane argmax per M row (16 candidates), tie -> lowest index
#pragma unroll
    for (int r = 0; r < 8; ++r) {
        const int mr = r + grp * 8;
        redS[wave][mr][m] = bs[r];
        redN[wave][mr][m] = bn[r];
    }
    __syncthreads();
    if (threadIdx.x < 64) {
        const int w  = threadIdx.x >> 4;
        const int mr = threadIdx.x & 15;
        float best = redS[w][mr][0];
        int   bId  = redN[w][mr][0];
        for (int c = 1; c < 16; ++c) {
            float sc = redS[w][mr][c];
            int   id = redN[w][mr][c];
            if (sc > best || (sc == best && id < bId)) { best = sc; bId = id; }
        }
        tokens[(blockIdx.x * 4 + w) * 16 + mr] = bId;
    }
}

// ---------------------------------------------------------------- upsample + update
__device__ __forceinline__ float keys_w(float t) {   // Keys cubic, a = -0.5
    t = fabsf(t);
    if (t <= 1.f) return (1.5f * t - 2.5f) * t * t + 1.f;
    if (t <  2.f) return ((-0.5f * t + 2.5f) * t - 4.f) * t + 2.f;
    return 0.f;
}

__global__ void upsample_update(const int* __restrict__ tokens,
                                const float* __restrict__ emb,
                                float* __restrict__ z_dec,
                                float* __restrict__ z_rest,
                                float* __restrict__ out, int ph) {
    int idx = blockIdx.x * blockDim.x + threadIdx.x;
    if (idx >= TOT) return;
    const int b = idx >> 13;
    const int c = (idx >> 8) & 31;
    const int y = (idx >> 4) & 15;
    const int x = idx & 15;

    const float scale = (float)H_ / (float)ph;       // >= 1 (no antialias)
    const float yin = ((float)y + 0.5f) / scale - 0.5f;
    const float xin = ((float)x + 0.5f) / scale - 0.5f;
    const int by = (int)floorf(yin);
    const int bx = (int)floorf(xin);

    float wy[4], wx[4], sumy = 0.f, sumx = 0.f;
#pragma unroll
    for (int i = 0; i < 4; ++i) {
        int p = by - 1 + i;
        float w = (p >= 0 && p < ph) ? keys_w(yin - (float)p) : 0.f;
        wy[i] = w; sumy += w;
        int q = bx - 1 + i;
        float v = (q >= 0 && q < ph) ? keys_w(xin - (float)q) : 0.f;
        wx[i] = v; sumx += v;
    }

    float accv = 0.f;
    const int tb = b * ph * ph;
#pragma unroll
    for (int i = 0; i < 4; ++i) {
        int p = by - 1 + i;
        if (p < 0 || p >= ph) continue;
        float rowacc = 0.f;
#pragma unroll
        for (int j = 0; j < 4; ++j) {
            int q = bx - 1 + j;
            if (q < 0 || q >= ph) continue;
            int t = tokens[tb + p * ph + q];
            rowacc += wx[j] * emb[t * C_ + c];
        }
        accv += wy[i] * rowacc;
    }
    const float zup = accv / (sumy * sumx);          // per-axis renormalization

    const float nd = z_dec[idx] + zup;
    z_dec[idx]  = nd;
    out[idx]    = nd;
    z_rest[idx] -= zup;
}

// ---------------------------------------------------------------- launch
extern "C" void kernel_launch(void* const* d_in, const int* in_sizes, int n_in,
                              void* d_out, int out_size, void* d_ws, size_t ws_size,
                              hipStream_t stream) {
    (void)in_sizes; (void)n_in; (void)out_size; (void)ws_size;
    const float* z_enc = (const float*)d_in[0];
    const float* emb   = (const float*)d_in[1];
    float* out    = (float*)d_out;

    float* z_rest = (float*)d_ws;          // 524288 f
    float* z_dec  = z_rest + TOT;          // 524288 f
    float* wsqh   = z_dec + TOT;           // 4096 f
    int*   tokens = (int*)(wsqh + K_);     // 16384 i

    init_state<<<TOT / 256, 256, 0, stream>>>(z_enc, z_rest, z_dec);
    wsq_half<<<K_ / 256, 256, 0, stream>>>(emb, wsqh);

    for (int level = 0; level < 5; ++level) {
        const int ph = 1 << level;               // 1,2,4,8,16
        const int R  = B_ * ph * ph;             // zflat rows (all %64 == 0)
        vq_argmin<<<R / 64, 128, 0, stream>>>(z_rest, emb, wsqh, tokens, ph);
        upsample_update<<<TOT / 256, 256, 0, stream>>>(
            tokens, emb, z_dec, z_rest, out + (size_t)level * TOT, ph);
    }
}